// GlobalModel_49546742726709
// MI455X (gfx1250) — compile-verified
//
#include <hip/hip_runtime.h>
#include <hip/hip_bf16.h>

typedef float v2f __attribute__((ext_vector_type(2)));
typedef float v8f __attribute__((ext_vector_type(8)));

#define N_GRAPHS 4096
#define HIDDEN   128
#define MLP_H    100
#define MLP_HP   112   // MLP_H padded to 7x16 tiles
#define LN_EPS   1e-5f

// ---------------------------------------------------------------------------
// Kernel A: zero the aggregation buffer (4096 x 128 f32 = 2 MB in d_ws)
// ---------------------------------------------------------------------------
__global__ void __launch_bounds__(256) zero_agg_kernel(float4* __restrict__ agg4) {
    agg4[(size_t)blockIdx.x * 256 + threadIdx.x] = make_float4(0.f, 0.f, 0.f, 0.f);
}

// ---------------------------------------------------------------------------
// Kernel A2: repack weights into zero-padded, fragment-native layout
//   Wp[kt][c][kk] = W[kt*4+kk][c]  (0 outside [krows, ncols]),
// so a B fragment is ONE unconditional global_load_b64 per lane.
// ---------------------------------------------------------------------------
__global__ void __launch_bounds__(256) pad_w_kernel(const float* __restrict__ W,
                                                    float* __restrict__ Wp,
                                                    int krows, int ncols,
                                                    int ktiles, int ncp) {
    int idx = blockIdx.x * 256 + threadIdx.x;
    int total = ktiles * ncp * 4;
    if (idx >= total) return;
    int kk = idx & 3;
    int c  = (idx >> 2) % ncp;
    int kt = (idx >> 2) / ncp;
    int k  = kt * 4 + kk;
    Wp[idx] = (k < krows && c < ncols) ? W[(size_t)k * ncols + c] : 0.0f;
}

__global__ void __launch_bounds__(128) pad_b_kernel(const float* __restrict__ b,
                                                    float* __restrict__ bp,
                                                    int n, int np) {
    int i = threadIdx.x;
    if (i < np) bp[i] = (i < n) ? b[i] : 0.0f;
}

// ---------------------------------------------------------------------------
// Kernel B: segment_sum of x[2M,128] by sorted batch ids into agg[4096,128].
// 1.024 GB streaming read -> the HBM-bound hot loop (~44us roofline).
// Each wave: 64 contiguous nodes; each lane: 4 columns via float4.
// Register accumulation while graph id is constant; atomic flush on change.
// ---------------------------------------------------------------------------
__global__ void __launch_bounds__(256) seg_sum_kernel(const float* __restrict__ x,
                                                      const int* __restrict__ batch,
                                                      float* __restrict__ agg,
                                                      int n_nodes) {
    const int lane = threadIdx.x & 31;
    const int wave = threadIdx.x >> 5;
    int base = blockIdx.x * 512 + wave * 64;
    if (base >= n_nodes) return;
    int end = base + 64;
    if (end > n_nodes) end = n_nodes;

    const float4* __restrict__ x4 = (const float4*)x;
    int cur = batch[base];
    float4 acc = make_float4(0.f, 0.f, 0.f, 0.f);

    for (int node = base; node < end; ++node) {
        int g = batch[node];                       // wave-uniform load
        if (g != cur) {
            float* dst = agg + (size_t)cur * HIDDEN + lane * 4;
            atomicAdd(dst + 0, acc.x);
            atomicAdd(dst + 1, acc.y);
            atomicAdd(dst + 2, acc.z);
            atomicAdd(dst + 3, acc.w);
            acc = make_float4(0.f, 0.f, 0.f, 0.f);
            cur = g;
        }
        float4 v = x4[(size_t)node * 32 + lane];   // coalesced 512B/wave
        acc.x += v.x; acc.y += v.y; acc.z += v.z; acc.w += v.w;
    }
    float* dst = agg + (size_t)cur * HIDDEN + lane * 4;
    atomicAdd(dst + 0, acc.x);
    atomicAdd(dst + 1, acc.y);
    atomicAdd(dst + 2, acc.z);
    atomicAdd(dst + 3, acc.w);
}

// ---------------------------------------------------------------------------
// WMMA f32 16x16x4 fragment helpers (layouts per CDNA5 ISA 7.12.2)
// ---------------------------------------------------------------------------
// A 16x4 (MxK), row-major source: lanes 0-15 hold K=0/1, lanes 16-31 K=2/3.
// Two consecutive floats -> single ds_load_b64.
__device__ __forceinline__ v2f load_a_frag(const float* base, int ld, int lane) {
    int m  = lane & 15;
    int kb = (lane >> 4) << 1;      // 0 or 2
    return *(const v2f*)(base + m * ld + kb);
}

// C/D 16x16: VGPR r -> M=r (lanes 0-15, N=lane) / M=r+8 (lanes 16-31)
__device__ __forceinline__ void store_c_frag(float* base, int ld, v8f c, int lane) {
    int n  = lane & 15;
    int mo = (lane >> 4) << 3;      // 0 or 8
#pragma unroll
    for (int r = 0; r < 8; ++r) base[(size_t)(r + mo) * ld + n] = c[r];
}

// One 16-row GEMM layer: H_out = act(A[16,K] * Wp + biasp), all padded,
// branch-free inner loop: ds_load_b64 (A) + global_load_b64 (B) + wmma.
__device__ __forceinline__ void gemm16_layer(const float* Alds, int lda, int ktiles,
                                             const float* __restrict__ Wp, int ncp,
                                             const float* __restrict__ biasp, bool relu,
                                             float* Hout, int ldo, int ncol_tiles,
                                             int lane) {
    const int n  = lane & 15;
    const int kb = (lane >> 4) << 1;                 // 0 or 2
    for (int ct = 0; ct < ncol_tiles; ++ct) {
        const int col0 = ct * 16;
        const float* wbase = Wp + (size_t)(col0 + n) * 4 + kb;  // advance by ncp*4/kt
        v8f acc = {};
#pragma unroll 4
        for (int kt = 0; kt < ktiles; ++kt) {
            v2f a = load_a_frag(Alds + kt * 4, lda, lane);
            v2f b = *(const v2f*)(wbase + (size_t)kt * ncp * 4);
            acc = __builtin_amdgcn_wmma_f32_16x16x4_f32(
                /*neg_a=*/false, a, /*neg_b=*/false, b,
                /*c_mod=*/(short)0, acc, /*reuse_a=*/false, /*reuse_b=*/false);
        }
        float bv = biasp[col0 + n];
#pragma unroll
        for (int r = 0; r < 8; ++r) {
            float v = acc[r] + bv;
            if (relu) v = fmaxf(v, 0.0f);
            acc[r] = v;
        }
        store_c_frag(Hout + col0, ldo, acc, lane);
    }
}

// ---------------------------------------------------------------------------
// Kernel C: per 16-row tile: concat(u,agg) -> 4 WMMA GEMM layers -> LN -> +u
// One wave (32 threads) per block; 256 blocks cover all 4096 graphs.
// ---------------------------------------------------------------------------
__global__ void __launch_bounds__(32) mlp_ln_kernel(
    const float* __restrict__ u, const float* __restrict__ agg,
    const float* __restrict__ W1p, const float* __restrict__ b1p,
    const float* __restrict__ W2p, const float* __restrict__ b2p,
    const float* __restrict__ W3p, const float* __restrict__ b3p,
    const float* __restrict__ W4p, const float* __restrict__ b4p,
    const float* __restrict__ gamma, const float* __restrict__ beta,
    float* __restrict__ out) {

    __shared__ float IN[16 * 256];     // concat(u, agg) tile
    __shared__ float HA[16 * MLP_HP];  // hidden ping
    __shared__ float HB[16 * MLP_HP];  // hidden pong
    __shared__ float H4[16 * HIDDEN];  // pre-LN output

    const int lane = threadIdx.x;        // 0..31
    const int row0 = blockIdx.x * 16;

    // Stage concat(u, agg): 32 lanes x float4 == 128 cols per row
    const float4* __restrict__ u4 = (const float4*)u;
    const float4* __restrict__ a4 = (const float4*)agg;
    float4* IN4 = (float4*)IN;
#pragma unroll 4
    for (int r = 0; r < 16; ++r) {
        IN4[r * 64 + lane]      = u4[(size_t)(row0 + r) * 32 + lane];
        IN4[r * 64 + 32 + lane] = a4[(size_t)(row0 + r) * 32 + lane];
    }
    __syncthreads();

    // Layer 1: [16,256] x W1[256,100] -> HA[16,112]  (64 k-steps, 7 col tiles)
    gemm16_layer(IN, 256, 64, W1p, MLP_HP, b1p, true, HA, MLP_HP, 7, lane);
    __syncthreads();
    // Layer 2: [16,112] x W2 -> HB (k rows 100..111 are zero-padded in W2p)
    gemm16_layer(HA, MLP_HP, 28, W2p, MLP_HP, b2p, true, HB, MLP_HP, 7, lane);
    __syncthreads();
    // Layer 3
    gemm16_layer(HB, MLP_HP, 28, W3p, MLP_HP, b3p, true, HA, MLP_HP, 7, lane);
    __syncthreads();
    // Layer 4: [16,112] x W4[100,128] -> H4[16,128], no relu
    gemm16_layer(HA, MLP_HP, 28, W4p, HIDDEN, b4p, false, H4, HIDDEN, 8, lane);
    __syncthreads();

    // LayerNorm + residual. Lane pair (r, r+16) splits row r's 128 cols.
    const int r    = lane & 15;
    const int half = lane >> 4;
    const float* hrow = H4 + r * HIDDEN + half * 64;
    float s = 0.f, s2 = 0.f;
#pragma unroll 8
    for (int c = 0; c < 64; ++c) {
        float v = hrow[c];
        s += v; s2 += v * v;
    }
    s  += __shfl_xor(s, 16, 32);
    s2 += __shfl_xor(s2, 16, 32);
    const float mu  = s * (1.0f / HIDDEN);
    const float var = s2 * (1.0f / HIDDEN) - mu * mu;
    const float inv = rsqrtf(var + LN_EPS);

#pragma unroll 8
    for (int c = 0; c < 64; ++c) {
        int col = half * 64 + c;
        float h = H4[r * HIDDEN + col];
        out[(size_t)(row0 + r) * HIDDEN + col] =
            u[(size_t)(row0 + r) * HIDDEN + col] + (h - mu) * inv * gamma[col] + beta[col];
    }
}

// ---------------------------------------------------------------------------
// Launch
// ---------------------------------------------------------------------------
extern "C" void kernel_launch(void* const* d_in, const int* in_sizes, int n_in,
                              void* d_out, int out_size, void* d_ws, size_t ws_size,
                              hipStream_t stream) {
    const float* x     = (const float*)d_in[0];
    // d_in[1] edge_index, d_in[2] edge_attr: unused by the global model
    const float* u     = (const float*)d_in[3];
    const int*   batch = (const int*)d_in[4];
    const float* W1    = (const float*)d_in[5];
    const float* b1    = (const float*)d_in[6];
    const float* W2    = (const float*)d_in[7];
    const float* b2    = (const float*)d_in[8];
    const float* W3    = (const float*)d_in[9];
    const float* b3    = (const float*)d_in[10];
    const float* W4    = (const float*)d_in[11];
    const float* b4    = (const float*)d_in[12];
    const float* gam   = (const float*)d_in[13];
    const float* bet   = (const float*)d_in[14];
    float* out = (float*)d_out;
    const int n_nodes = in_sizes[4];              // batch element count

    // Workspace layout (floats)
    float* ws  = (float*)d_ws;
    float* agg = ws;                              // 4096*128         = 524288
    float* W1p = agg + (size_t)N_GRAPHS * HIDDEN; // 64*112*4         =  28672
    float* W2p = W1p + 64 * MLP_HP * 4;           // 28*112*4         =  12544
    float* W3p = W2p + 28 * MLP_HP * 4;           // 28*112*4         =  12544
    float* W4p = W3p + 28 * MLP_HP * 4;           // 28*128*4         =  14336
    float* b1p = W4p + 28 * HIDDEN * 4;           // 112
    float* b2p = b1p + MLP_HP;                    // 112
    float* b3p = b2p + MLP_HP;                    // 112
    float* b4p = b3p + MLP_HP;                    // 128

    // A: zero agg (131072 float4 -> 512 blocks x 256)
    zero_agg_kernel<<<(N_GRAPHS * HIDDEN / 4 + 255) / 256, 256, 0, stream>>>((float4*)agg);

    // A2: repack weights/biases into padded fragment-native layout
    pad_w_kernel<<<(64 * MLP_HP * 4 + 255) / 256, 256, 0, stream>>>(W1, W1p, 256,   MLP_H,  64, MLP_HP);
    pad_w_kernel<<<(28 * MLP_HP * 4 + 255) / 256, 256, 0, stream>>>(W2, W2p, MLP_H, MLP_H,  28, MLP_HP);
    pad_w_kernel<<<(28 * MLP_HP * 4 + 255) / 256, 256, 0, stream>>>(W3, W3p, MLP_H, MLP_H,  28, MLP_HP);
    pad_w_kernel<<<(28 * HIDDEN * 4 + 255) / 256, 256, 0, stream>>>(W4, W4p, MLP_H, HIDDEN, 28, HIDDEN);
    pad_b_kernel<<<1, 128, 0, stream>>>(b1, b1p, MLP_H, MLP_HP);
    pad_b_kernel<<<1, 128, 0, stream>>>(b2, b2p, MLP_H, MLP_HP);
    pad_b_kernel<<<1, 128, 0, stream>>>(b3, b3p, MLP_H, MLP_HP);
    pad_b_kernel<<<1, 128, 0, stream>>>(b4, b4p, HIDDEN, HIDDEN);

    // B: streaming segment-sum (512 nodes per 256-thread block)
    seg_sum_kernel<<<(n_nodes + 511) / 512, 256, 0, stream>>>(x, batch, agg, n_nodes);

    // C: WMMA MLP + LayerNorm + residual (one wave per 16 rows)
    mlp_ln_kernel<<<N_GRAPHS / 16, 32, 0, stream>>>(u, agg, W1p, b1p, W2p, b2p,
                                                    W3p, b3p, W4p, b4p, gam, bet, out);
}